// Transducer_44470091383515
// MI455X (gfx1250) — compile-verified
//
#include <hip/hip_runtime.h>
#include <math.h>

// ---------------------------------------------------------------------------
// Pruned RNN-T lattice loss for MI455X (gfx1250, wave32).
//
// Roofline: phase 1 streams 32*512*5*512 f32 = 167.8 MB of logits once
// (~7.2 us at 23.3 TB/s). We reduce each 512-float row to (em, bl) on the
// fly (640 KB of intermediates in d_ws), so HBM traffic is ~1x the input.
// Phase 2 is a single-wave serial scan (lane == sequence, N == 32 == wave32)
// optimized for latency: flattened prefix-LSE chain + software-pipelined,
// double-buffered loads (511 = 73 blocks of 7 timesteps).
//
// CDNA5 path: the 32-lane sum-of-exp reduction in phase 1 uses
// v_wmma_f32_16x16x4_f32 (f32 matrix pipe) instead of a shuffle tree.
//
// Transcendentals use raw v_exp_f32 / v_log_f32 (no OCML denorm guards);
// exp arguments are formed with a single FMA: exp(x-m) = exp2(fma(x,log2e,c))
// with c = -m*log2e hoisted out of the 16-element sum.
// ---------------------------------------------------------------------------

#define NEGF (-1e30f)
#define K_LOG2E 1.44269504088896340736f
#define K_LN2   0.69314718055994530942f

constexpr int N = 32, T = 512, S = 5, C = 512, U = 256;
constexpr int ROWS = N * T * S;        // 81920 softmax rows
constexpr int UF = 7;                  // timestep unroll (511 = 73 * 7)
constexpr int NB = (T - 1) / UF;       // 73 blocks

typedef float v2f __attribute__((ext_vector_type(2)));
typedef float v8f __attribute__((ext_vector_type(8)));

// Raw hardware transcendentals: single v_exp_f32 / v_log_f32.
__device__ __forceinline__ float rexp2(float x) {
  return __builtin_amdgcn_exp2f(x);
}
__device__ __forceinline__ float rlog2(float x) {
  return __builtin_amdgcn_logf(x);
}
// exp(x - m) with c = -m*log2e precomputed: one v_fma + one v_exp.
__device__ __forceinline__ float expsub(float x, float c) {
  return rexp2(fmaf(x, K_LOG2E, c));
}

// ---------------------------------------------------------------------------
// Kernel 1: per-row logsumexp + gather of em/bl.
// One wave per row of 512 floats; lane loads 4x float4 (fully coalesced:
// each b128 load covers 512 contiguous bytes across the wave).
// ---------------------------------------------------------------------------
__global__ __launch_bounds__(256) void k_softmax_em_bl(
    const float* __restrict__ logits, const int* __restrict__ ranges,
    const int* __restrict__ y, float* __restrict__ em_ws,
    float* __restrict__ bl_ws) {
  const int wave = threadIdx.x >> 5;
  const int lane = threadIdx.x & 31;
  const int row  = blockIdx.x * 8 + wave;          // grid sized exactly

  const float4* rp = (const float4*)(logits + (size_t)row * C);
  float4 f0 = rp[lane];
  float4 f1 = rp[lane + 32];
  float4 f2 = rp[lane + 64];
  float4 f3 = rp[lane + 96];

  // per-lane max over 16 elements
  float m = fmaxf(fmaxf(fmaxf(f0.x, f0.y), fmaxf(f0.z, f0.w)),
                  fmaxf(fmaxf(f1.x, f1.y), fmaxf(f1.z, f1.w)));
  m = fmaxf(m, fmaxf(fmaxf(f2.x, f2.y), fmaxf(f2.z, f2.w)));
  m = fmaxf(m, fmaxf(fmaxf(f3.x, f3.y), fmaxf(f3.z, f3.w)));
  // wave max (5-step xor tree)
#pragma unroll
  for (int i = 1; i < 32; i <<= 1) m = fmaxf(m, __shfl_xor(m, i, 32));

  // per-lane sum of exp(x - m): 16 independent v_fma + v_exp_f32 pairs
  const float cm = -m * K_LOG2E;
  float s = expsub(f0.x, cm) + expsub(f0.y, cm) +
            expsub(f0.z, cm) + expsub(f0.w, cm) +
            expsub(f1.x, cm) + expsub(f1.y, cm) +
            expsub(f1.z, cm) + expsub(f1.w, cm) +
            expsub(f2.x, cm) + expsub(f2.y, cm) +
            expsub(f2.z, cm) + expsub(f2.w, cm) +
            expsub(f3.x, cm) + expsub(f3.y, cm) +
            expsub(f3.z, cm) + expsub(f3.w, cm);

  // Cross-lane sum via f32 WMMA (EXEC all-ones here; uniform control flow).
  // A 16x4 f32 layout: lanes 0-15 VGPR0 = A[m][0], lanes 16-31 VGPR0 = A[m][2].
  // With a = {s, 0} and B = ones: D[m][n] = s(m) + s(m+16) for every n.
  v2f a; a.x = s;    a.y = 0.0f;
  v2f b; b.x = 1.0f; b.y = 1.0f;
  v8f c = {0.f, 0.f, 0.f, 0.f, 0.f, 0.f, 0.f, 0.f};
  v8f d = __builtin_amdgcn_wmma_f32_16x16x4_f32(
      /*neg_a=*/false, a, /*neg_b=*/false, b,
      /*c_mod=*/(short)0, c, /*reuse_a=*/false, /*reuse_b=*/false);
  // lanes 0-15 hold rows 0-7 (pair sums), lanes 16-31 hold rows 8-15
  float tot = d[0] + d[1] + d[2] + d[3] + d[4] + d[5] + d[6] + d[7];
  tot += __shfl_xor(tot, 16, 32);      // full 32-lane sum on every lane

  float lse = m + K_LN2 * rlog2(tot);  // tot in [1, 512]: raw log is exact

  if (lane == 0) {
    const int n  = row / (T * S);
    const int ts = row % (T * S);
    const int r  = ranges[row];                  // flat (n,t,s) == row
    const int sym = y[n * U + r];
    em_ws[ts * N + n] = logits[(size_t)row * C + sym] - lse;  // L0/L2 hot
    bl_ws[ts * N + n] = f0.x - lse;              // lane 0 holds element 0
  }
}

// ---------------------------------------------------------------------------
// Kernel 2: lattice recursion. One wave, lane == sequence n (N == 32).
// ---------------------------------------------------------------------------

// Flattened emit chain (LSE semiring):
//   a'[s] = P[s] + LSE_{j<=s}(a[j] - P[j]),  P[s] = e[0]+...+e[s-1]
// All exps issue in parallel; one exp->log latency on the critical path.
__device__ __forceinline__ void chain_lse5(float a[5], const float e[5]) {
  float P1 = e[0], P2 = P1 + e[1], P3 = P2 + e[2], P4 = P3 + e[3];
  float b0 = a[0], b1 = a[1] - P1, b2 = a[2] - P2, b3 = a[3] - P3,
        b4 = a[4] - P4;
  float Mg = fmaxf(fmaxf(fmaxf(b0, b1), fmaxf(b2, b3)), b4);
  Mg = fmaxf(Mg, -1e37f);              // guard against all -inf
  const float cm = -Mg * K_LOG2E;      // fold (b-Mg)*log2e into one FMA
  float x0 = expsub(b0, cm);
  float x1 = expsub(b1, cm);
  float x2 = expsub(b2, cm);
  float x3 = expsub(b3, cm);
  float x4 = expsub(b4, cm);
  float X1 = x0 + x1, X2 = X1 + x2, X3 = X2 + x3, X4 = X3 + x4;
  a[1] = P1 + Mg + K_LN2 * rlog2(X1);
  a[2] = P2 + Mg + K_LN2 * rlog2(X2);
  a[3] = P3 + Mg + K_LN2 * rlog2(X3);
  a[4] = P4 + Mg + K_LN2 * rlog2(X4);
}

// Same flattening, max (Viterbi) semiring.
__device__ __forceinline__ void chain_max5(float a[5], const float e[5]) {
  float P1 = e[0], P2 = P1 + e[1], P3 = P2 + e[2], P4 = P3 + e[3];
  float r = a[0];
  r = fmaxf(r, a[1] - P1); a[1] = P1 + r;
  r = fmaxf(r, a[2] - P2); a[2] = P2 + r;
  r = fmaxf(r, a[3] - P3); a[3] = P3 + r;
  r = fmaxf(r, a[4] - P4); a[4] = P4 + r;
}

__device__ __forceinline__ float pick5(const float g[5], int src) {
  float r = NEGF;                       // src >= S -> pruned (NEG)
  r = (src == 0) ? g[0] : r;
  r = (src == 1) ? g[1] : r;
  r = (src == 2) ? g[2] : r;
  r = (src == 3) ? g[3] : r;
  r = (src == 4) ? g[4] : r;
  return r;
}

__device__ __forceinline__ void load_block(
    int b0, int n, const float* __restrict__ em_ws,
    const float* __restrict__ bl_ws, const int* __restrict__ ranges,
    float e[UF][5], float bb[UF][5], int dl[UF]) {
  int lb[UF + 1];
#pragma unroll
  for (int i = 0; i <= UF; ++i) lb[i] = ranges[n * T * S + (b0 + i) * S];
#pragma unroll
  for (int i = 0; i < UF; ++i) dl[i] = lb[i + 1] - lb[i];
#pragma unroll
  for (int i = 0; i < UF; ++i) {
#pragma unroll
    for (int s = 0; s < 5; ++s) {
      e[i][s]  = em_ws[((b0 + i) * S + s) * N + n];   // coalesced across wave
      bb[i][s] = bl_ws[((b0 + i) * S + s) * N + n];
    }
  }
}

__global__ __launch_bounds__(32) void k_lattice(
    const float* __restrict__ em_ws, const float* __restrict__ bl_ws,
    const int* __restrict__ ranges, float* __restrict__ out) {
  const int n = threadIdx.x;            // lane == sequence, N == 32

  float aL[5], aM[5];
  aL[0] = 0.0f; aM[0] = 0.0f;
#pragma unroll
  for (int s = 1; s < 5; ++s) { aL[s] = NEGF; aM[s] = NEGF; }

  float eC[UF][5], bC[UF][5]; int dC[UF];
  load_block(0, n, em_ws, bl_ws, ranges, eC, bC, dC);

  for (int blk = 0; blk < NB; ++blk) {
    // Prefetch next block while this block's serial recurrence runs.
    float eN[UF][5], bN[UF][5]; int dN[UF];
    const bool more = (blk + 1 < NB);
    if (more) load_block((blk + 1) * UF, n, em_ws, bl_ws, ranges, eN, bN, dN);

#pragma unroll
    for (int i = 0; i < UF; ++i) {
      chain_lse5(aL, eC[i]);
      chain_max5(aM, eC[i]);
      float gL[5], gM[5];
#pragma unroll
      for (int s = 0; s < 5; ++s) {
        gL[s] = aL[s] + bC[i][s];
        gM[s] = aM[s] + bC[i][s];
      }
      const int delta = dC[i];
#pragma unroll
      for (int s = 0; s < 5; ++s) {
        aL[s] = pick5(gL, s + delta);
        aM[s] = pick5(gM, s + delta);
      }
    }

    if (more) {
#pragma unroll
      for (int i = 0; i < UF; ++i) {
        dC[i] = dN[i];
#pragma unroll
        for (int s = 0; s < 5; ++s) { eC[i][s] = eN[i][s]; bC[i][s] = bN[i][s]; }
      }
    }
  }

  // Final emit chain at t = T-1 (no shift).
  float eF[5];
#pragma unroll
  for (int s = 0; s < 5; ++s) eF[s] = em_ws[((T - 1) * S + s) * N + n];
  chain_lse5(aL, eF);
  chain_max5(aM, eF);

  float vL = aL[4], vM = aM[4];
#pragma unroll
  for (int i = 1; i < 32; i <<= 1) {
    vL += __shfl_xor(vL, i, 32);
    vM += __shfl_xor(vM, i, 32);
  }
  if (n == 0) {
    out[0] = -vL;   // pruned (logsumexp) loss
    out[1] = -vM;   // one-best (Viterbi) loss
  }
}

// ---------------------------------------------------------------------------
extern "C" void kernel_launch(void* const* d_in, const int* in_sizes, int n_in,
                              void* d_out, int out_size, void* d_ws,
                              size_t ws_size, hipStream_t stream) {
  const float* logits = (const float*)d_in[0];
  const int*   ranges = (const int*)d_in[1];
  const int*   y      = (const int*)d_in[2];
  // d_in[3] (x_lens) is unused: all sequences are full length T.

  float* em_ws = (float*)d_ws;                 // [T*S][N] = 320 KB
  float* bl_ws = em_ws + (size_t)ROWS;         // [T*S][N] = 320 KB

  k_softmax_em_bl<<<ROWS / 8, 256, 0, stream>>>(logits, ranges, y, em_ws,
                                                bl_ws);
  k_lattice<<<1, 32, 0, stream>>>(em_ws, bl_ws, ranges, (float*)d_out);
}